// SeqGraphEncoder_26070451486835
// MI455X (gfx1250) — compile-verified
//
#include <hip/hip_runtime.h>
#include <hip/hip_bf16.h>

// ---------------------------------------------------------------------------
// SeqGraphEncoder for MI455X (gfx1250, wave32).
//
// Math reduction:  einsum('ed,fd,f->e', X, W, a) == X . (W^T a)
//   v_src = W^T alpha_src ; v_dst = W^T alpha_dst      (64-vectors)
//   p_src[n] = x[n].v_src ; p_dst[n] = x[n].v_dst      (per node, x = POI[sess])
//   q_src[k] = delta_dis[k].v_src                      (512 buckets)
//   fwd score(e) = p_src[dst[e]] + q_src[dist[e]], target dst[e]
//   bwd score(e) = p_dst[src[e]],                   target src[e]
// Then segment softmax over targets and H[t] += attn * x[source].
//
// WMMA: p/q projections done as P = X_tile(16x64) @ V(64x16) with
// V_WMMA_F32_16X16X4_F32 (full f32 precision), K in 16 chunks of 4.
// V is stored transposed+padded (Vt[16][64], rows 2..15 zero) so every lane's
// B operand is one unconditional 8-byte load — no divergent guarded loads.
// ---------------------------------------------------------------------------

typedef float v2f __attribute__((ext_vector_type(2)));
typedef float v8f __attribute__((ext_vector_type(8)));

#define D_DIM 64
#define ENC_NEG_INF 0x007FFFFFu   // order-preserving encoding of -inf

__device__ __forceinline__ unsigned enc_f32(float f) {
    unsigned u = __float_as_uint(f);
    return (u & 0x80000000u) ? ~u : (u | 0x80000000u);
}
__device__ __forceinline__ float dec_f32(unsigned u) {
    u = (u & 0x80000000u) ? (u & 0x7fffffffu) : ~u;
    return __uint_as_float(u);
}

// --------------------- Vt[16][64]: row0=W^T a_src, row1=W^T a_dst ----------
__global__ void compute_v_kernel(const float* __restrict__ W,
                                 const float* __restrict__ a_src,
                                 const float* __restrict__ a_dst,
                                 float* __restrict__ Vt) {
    int d = threadIdx.x;               // 64 threads
    float s0 = 0.f, s1 = 0.f;
    #pragma unroll 8
    for (int f = 0; f < D_DIM; ++f) {
        float w = W[f * D_DIM + d];
        s0 = fmaf(w, a_src[f], s0);
        s1 = fmaf(w, a_dst[f], s1);
    }
    Vt[d]         = s0;
    Vt[D_DIM + d] = s1;
    #pragma unroll
    for (int n = 2; n < 16; ++n) Vt[n * D_DIM + d] = 0.f;
}

// ------------------- WMMA projection: rows @ Vt^T --------------------------
// One wave handles 16 rows. Operand layouts (V_WMMA_F32_16X16X4_F32):
//   A: lane L (m=L&15, h=L>>4): a = { A[m][k0+2h], A[m][k0+2h+1] }
//   B: lane L (n=L&15, h=L>>4): b = { B[k0+2h][n], B[k0+2h+1][n] } = Vt[n][k0+2h..]
//   D: lane L (n=L&15, h=L>>4): acc[v] = D[v+8h][n]
__global__ void proj_rows_wmma_kernel(const float* __restrict__ table,
                                      const int* __restrict__ idx,   // nullptr -> identity
                                      const float* __restrict__ Vt,  // [16][64]
                                      float* __restrict__ pa,
                                      float* __restrict__ pb,
                                      int n_rows) {
    const int lane  = threadIdx.x & 31;
    const int waveg = (int)((blockIdx.x * blockDim.x + threadIdx.x) >> 5);
    const int base  = waveg * 16;
    if (base >= n_rows) return;        // wave-uniform: EXEC stays all-ones below

    const int m = lane & 15;
    const int h = lane >> 4;

    int r = base + m;
    if (r >= n_rows) r = n_rows - 1;   // per-lane clamp; masked at store
    const int row = idx ? idx[r] : r;
    const v2f* __restrict__ arow = (const v2f*)(table + (long long)row * D_DIM);
    const v2f* __restrict__ brow = (const v2f*)(Vt + m * D_DIM);

    v8f acc = {};
    #pragma unroll
    for (int kk = 0; kk < 16; ++kk) {
        v2f a = arow[2 * kk + h];      // one global_load_b64
        v2f b = brow[2 * kk + h];      // one global_load_b64, no divergence
        acc = __builtin_amdgcn_wmma_f32_16x16x4_f32(
            /*neg_a=*/false, a, /*neg_b=*/false, b,
            /*c_mod=*/(short)0, acc, /*reuse_a=*/false, /*reuse_b=*/false);
    }

    // Column 0 of D = p_a, column 1 = p_b.  Lane n==0/1 (h selects M half).
    if (m <= 1) {
        float* __restrict__ op = (m == 0) ? pa : pb;
        #pragma unroll
        for (int v = 0; v < 8; ++v) {
            int mm = base + v + 8 * h;
            if (mm < n_rows) op[mm] = acc[v];
        }
    }
}

// ------------------------------- init --------------------------------------
__global__ void init_kernel(float* __restrict__ out, unsigned* __restrict__ m_enc,
                            float* __restrict__ s, int n_nodes) {
    long long gid = (long long)blockIdx.x * blockDim.x + threadIdx.x;
    long long tot = (long long)n_nodes * D_DIM;
    if (gid < tot) out[gid] = 0.f;
    if (gid < n_nodes) { m_enc[gid] = ENC_NEG_INF; s[gid] = 0.f; }
}

// --------------------------- pass 1: segment max ---------------------------
__global__ void edge_max_kernel(const int* __restrict__ srcp, const int* __restrict__ dstp,
                                const int* __restrict__ dist,
                                const float* __restrict__ p_src, const float* __restrict__ p_dst,
                                const float* __restrict__ q_src,
                                unsigned* __restrict__ m_enc, int E) {
    long long e = (long long)blockIdx.x * blockDim.x + threadIdx.x;
    if (e >= 2LL * E) return;
    __builtin_prefetch(dstp + e + 4096, 0, 1);   // speculative: global_prefetch_b8
    int t; float score;
    if (e < E) { t = dstp[e];     score = p_src[t] + q_src[dist[e]]; }
    else       { t = srcp[e - E]; score = p_dst[t]; }
    atomicMax(&m_enc[t], enc_f32(score));
}

// ----------------- pass 2: exp weights + segment sum -----------------------
__global__ void edge_exp_kernel(const int* __restrict__ srcp, const int* __restrict__ dstp,
                                const int* __restrict__ dist,
                                const float* __restrict__ p_src, const float* __restrict__ p_dst,
                                const float* __restrict__ q_src,
                                const unsigned* __restrict__ m_enc,
                                float* __restrict__ s, float* __restrict__ wbuf, int E) {
    long long e = (long long)blockIdx.x * blockDim.x + threadIdx.x;
    if (e >= 2LL * E) return;
    __builtin_prefetch(dstp + e + 4096, 0, 1);
    int t; float score;
    if (e < E) { t = dstp[e];     score = p_src[t] + q_src[dist[e]]; }
    else       { t = srcp[e - E]; score = p_dst[t]; }
    float w = __expf(score - dec_f32(m_enc[t]));
    wbuf[e] = w;
    atomicAdd(&s[t], w);
}

// -------- pass 3: H[t] += w * x[source]; 16 threads/edge, float4 each ------
__global__ void edge_accum_kernel(const int* __restrict__ srcp, const int* __restrict__ dstp,
                                  const int* __restrict__ sess_idx,
                                  const float* __restrict__ POI,
                                  const float* __restrict__ wbuf,
                                  float* __restrict__ out, int E) {
    long long gid = (long long)blockIdx.x * blockDim.x + threadIdx.x;
    long long tot = 2LL * E * 16;
    if (gid >= tot) return;
    long long e   = gid >> 4;
    int part      = (int)(gid & 15);
    int sN, t;
    if (e < E) { sN = srcp[e];     t = dstp[e]; }
    else       { sN = dstp[e - E]; t = srcp[e - E]; }
    float w = wbuf[e];
    const float4* __restrict__ xr =
        (const float4*)(POI + (long long)sess_idx[sN] * D_DIM);
    float4 xv = xr[part];
    float* __restrict__ o = out + (long long)t * D_DIM + part * 4;
    atomicAdd(o + 0, w * xv.x);
    atomicAdd(o + 1, w * xv.y);
    atomicAdd(o + 2, w * xv.z);
    atomicAdd(o + 3, w * xv.w);
}

// ------------------------------ normalize ----------------------------------
__global__ void normalize_kernel(float* __restrict__ out, const float* __restrict__ s,
                                 int n_nodes) {
    long long gid = (long long)blockIdx.x * blockDim.x + threadIdx.x;
    long long tot = (long long)n_nodes * D_DIM;
    if (gid >= tot) return;
    out[gid] = out[gid] / (s[gid >> 6] + 1e-16f);
}

// ---------------------------------------------------------------------------
extern "C" void kernel_launch(void* const* d_in, const int* in_sizes, int n_in,
                              void* d_out, int out_size, void* d_ws, size_t ws_size,
                              hipStream_t stream) {
    const float* POI   = (const float*)d_in[0];   // [N_POI, 64]
    const float* dtab  = (const float*)d_in[1];   // [N_DIST, 64]
    const float* W     = (const float*)d_in[2];   // [64, 64]
    const float* a_src = (const float*)d_in[3];   // [64]
    const float* a_dst = (const float*)d_in[4];   // [64]
    const int*   sess  = (const int*)d_in[5];     // [N_NODES]
    const int*   eidx  = (const int*)d_in[6];     // [2, E]
    const int*   edist = (const int*)d_in[7];     // [E]

    const int n_nodes = in_sizes[5];
    const int E       = in_sizes[6] / 2;
    const int n_dist  = in_sizes[1] / D_DIM;
    const int* srcp = eidx;
    const int* dstp = eidx + E;
    float* out = (float*)d_out;

    // workspace carve-out (256B aligned slices)
    char* wsp = (char*)d_ws;
    auto carve = [&](size_t bytes) -> void* {
        void* p = (void*)wsp;
        wsp += (bytes + 255) & ~(size_t)255;
        return p;
    };
    float*    Vt    = (float*)carve(16 * D_DIM * sizeof(float));       // padded V^T
    float*    q_src = (float*)carve((size_t)n_dist * sizeof(float));
    float*    q_dst = (float*)carve((size_t)n_dist * sizeof(float));   // computed, unused
    float*    p_src = (float*)carve((size_t)n_nodes * sizeof(float));
    float*    p_dst = (float*)carve((size_t)n_nodes * sizeof(float));
    unsigned* m_enc = (unsigned*)carve((size_t)n_nodes * sizeof(unsigned));
    float*    ssum  = (float*)carve((size_t)n_nodes * sizeof(float));
    float*    wbuf  = (float*)carve(2ULL * E * sizeof(float));

    const int T = 256;

    // 1) Vt = padded [W^T a_src ; W^T a_dst ; 0...] (tiny)
    compute_v_kernel<<<1, D_DIM, 0, stream>>>(W, a_src, a_dst, Vt);

    // 2) WMMA projections: q over distance table (identity idx), p over gathered nodes
    {
        int rows_per_block = (T / 32) * 16;   // 128
        int gb = (n_dist + rows_per_block - 1) / rows_per_block;
        proj_rows_wmma_kernel<<<gb, T, 0, stream>>>(dtab, nullptr, Vt,
                                                    q_src, q_dst, n_dist);
        gb = (n_nodes + rows_per_block - 1) / rows_per_block;
        proj_rows_wmma_kernel<<<gb, T, 0, stream>>>(POI, sess, Vt,
                                                    p_src, p_dst, n_nodes);
    }

    // 3) init accumulators
    {
        long long tot = (long long)n_nodes * D_DIM;
        int gb = (int)((tot + T - 1) / T);
        init_kernel<<<gb, T, 0, stream>>>(out, m_enc, ssum, n_nodes);
    }

    // 4) segment max over 2E edges
    {
        long long tot = 2LL * E;
        int gb = (int)((tot + T - 1) / T);
        edge_max_kernel<<<gb, T, 0, stream>>>(srcp, dstp, edist, p_src, p_dst, q_src,
                                              m_enc, E);
    }

    // 5) exp weights + segment sum
    {
        long long tot = 2LL * E;
        int gb = (int)((tot + T - 1) / T);
        edge_exp_kernel<<<gb, T, 0, stream>>>(srcp, dstp, edist, p_src, p_dst, q_src,
                                              m_enc, ssum, wbuf, E);
    }

    // 6) weighted scatter-accumulate of source features
    {
        long long tot = 2LL * E * 16;
        int gb = (int)((tot + T - 1) / T);
        edge_accum_kernel<<<gb, T, 0, stream>>>(srcp, dstp, sess, POI, wbuf, out, E);
    }

    // 7) normalize by (segment sum + 1e-16)
    {
        long long tot = (long long)n_nodes * D_DIM;
        int gb = (int)((tot + T - 1) / T);
        normalize_kernel<<<gb, T, 0, stream>>>(out, ssum, n_nodes);
    }
}